// FLaGPENodeEncoder_73770358276664
// MI455X (gfx1250) — compile-verified
//
#include <hip/hip_runtime.h>
#include <hip/hip_bf16.h>

#define G_     16
#define N_     512
#define E_     4096
#define KHOP   16
#define DIMIN  64
#define DIMPE  28
#define HID    64
#define DIMX   100
#define LN_EPS 1e-5f

typedef float v2f __attribute__((ext_vector_type(2)));
typedef float v8f __attribute__((ext_vector_type(8)));

// ---------------------------------------------------------------- zero P
__global__ void zero_kernel(float4* __restrict__ p) {
    p[(size_t)blockIdx.x * blockDim.x + threadIdx.x] = make_float4(0.f, 0.f, 0.f, 0.f);
}

// ------------------------------------------------- scatter edges into adj
__global__ void scatter_kernel(const int* __restrict__ ei, float* __restrict__ P) {
    int idx = blockIdx.x * blockDim.x + threadIdx.x;   // G*E total
    if (idx >= G_ * E_) return;
    int g = idx / E_, e = idx % E_;
    int src = ei[(size_t)g * 2 * E_ + e];
    int dst = ei[(size_t)g * 2 * E_ + E_ + e];
    atomicAdd(&P[((size_t)g * N_ + src) * N_ + dst], 1.0f);
}

// ---------------------------------------------------- row-normalize: D^-1 A
__global__ void rownorm_kernel(float* __restrict__ P) {
    int row = blockIdx.x * blockDim.x + threadIdx.x;   // G*N rows
    float4* r = (float4*)(P + (size_t)row * N_);
    float s = 0.f;
    for (int j = 0; j < N_ / 4; ++j) { float4 v = r[j]; s += v.x + v.y + v.z + v.w; }
    float inv = 1.0f / fmaxf(s, 1.0f);
    for (int j = 0; j < N_ / 4; ++j) {
        float4 v = r[j];
        v.x *= inv; v.y *= inv; v.z *= inv; v.w *= inv;
        r[j] = v;
    }
}

// ---------------------------------------------------- feat[:,0] = (1-a)/N
__global__ void feat0_kernel(const float* __restrict__ alpha, float* __restrict__ feat) {
    int row = blockIdx.x * blockDim.x + threadIdx.x;
    float a = 1.0f / (1.0f + __expf(-alpha[0]));
    feat[(size_t)row * KHOP] = (1.0f - a) * (1.0f / (float)N_);
}

// ------------- feat[:,k] = mean_j Pk[i,j] * (same_frag ? (1-a) : a)
__global__ void reduce_kernel(const float* __restrict__ Pk, const int* __restrict__ frag,
                              const float* __restrict__ alpha, float* __restrict__ feat, int k) {
    __shared__ int sfrag[N_];
    int row = blockIdx.x * blockDim.x + threadIdx.x;   // 256 rows/block, same graph
    int g   = blockIdx.x >> 1;                         // (blockIdx*256)/512
    for (int j = threadIdx.x; j < N_; j += blockDim.x) sfrag[j] = frag[(size_t)g * N_ + j];
    __syncthreads();
    float a = 1.0f / (1.0f + __expf(-alpha[0]));
    float oma = 1.0f - a;
    int fi = sfrag[row & (N_ - 1)];
    const float* prow = Pk + (size_t)row * N_;
    float s = 0.f;
    for (int j = 0; j < N_; ++j)
        s += prow[j] * ((sfrag[j] == fi) ? oma : a);
    feat[(size_t)row * KHOP + k] = s * (1.0f / (float)N_);
}

// --------------------------------- fp32 WMMA GEMM: C = A @ B  (per-graph 512x512)
// grid = G*16 blocks of 128 threads; block tile = 128x128; wave tile = 64x64.
__global__ __launch_bounds__(128) void gemm_pow_kernel(const float* __restrict__ A,
                                                       const float* __restrict__ B,
                                                       float* __restrict__ C) {
    int g  = blockIdx.x >> 4;
    int t  = blockIdx.x & 15;
    int bm = (t >> 2) * 128;
    int bn = (t & 3) * 128;
    int wave = threadIdx.x >> 5;
    int lane = threadIdx.x & 31;
    int wm = bm + (wave >> 1) * 64;
    int wn = bn + (wave & 1) * 64;
    const float* Ag = A + (size_t)g * N_ * N_;
    const float* Bg = B + (size_t)g * N_ * N_;
    float*       Cg = C + (size_t)g * N_ * N_;

    int lm = lane & 15;            // M (for A) / N (for B) within 16-wide tile
    int kh = (lane >> 4) << 1;     // K sub-pair: lanes 0-15 -> {0,1}, lanes 16-31 -> {2,3}

    v8f acc[4][4];
    #pragma unroll
    for (int mt = 0; mt < 4; ++mt)
        #pragma unroll
        for (int nt = 0; nt < 4; ++nt)
            acc[mt][nt] = (v8f)0.0f;

    for (int kb = 0; kb < N_; kb += 4) {
        v2f afr[4], bfr[4];
        #pragma unroll
        for (int mt = 0; mt < 4; ++mt)
            afr[mt] = *(const v2f*)&Ag[(size_t)(wm + mt * 16 + lm) * N_ + kb + kh];
        #pragma unroll
        for (int nt = 0; nt < 4; ++nt) {
            int col = wn + nt * 16 + lm;
            v2f b;
            b.x = Bg[(size_t)(kb + kh) * N_ + col];
            b.y = Bg[(size_t)(kb + kh + 1) * N_ + col];
            bfr[nt] = b;
        }
        #pragma unroll
        for (int mt = 0; mt < 4; ++mt)
            #pragma unroll
            for (int nt = 0; nt < 4; ++nt)
                acc[mt][nt] = __builtin_amdgcn_wmma_f32_16x16x4_f32(
                    false, afr[mt], false, bfr[nt], (short)0, acc[mt][nt], false, false);
    }

    int rowoff = (lane >> 4) * 8;  // C layout: VGPR r -> rows r (lanes 0-15) and r+8 (16-31)
    int cc = lane & 15;
    #pragma unroll
    for (int mt = 0; mt < 4; ++mt)
        #pragma unroll
        for (int nt = 0; nt < 4; ++nt) {
            #pragma unroll
            for (int r = 0; r < 8; ++r)
                Cg[(size_t)(wm + mt * 16 + rowoff + r) * N_ + wn + nt * 16 + cc] = acc[mt][nt][r];
        }
}

// --------------- fused MLP (3x Linear+ReLU) + LayerNorm + x@Wx + concat
// 256 threads = 8 waves; one row per wave; 1024 blocks cover 8192 rows.
__global__ __launch_bounds__(256) void mlp_out_kernel(
    const float* __restrict__ feat, const float* __restrict__ x,
    const float* __restrict__ W1, const float* __restrict__ b1,
    const float* __restrict__ W2, const float* __restrict__ b2,
    const float* __restrict__ W3, const float* __restrict__ b3,
    const float* __restrict__ gamma, const float* __restrict__ beta,
    const float* __restrict__ Wx, const float* __restrict__ bx,
    float* __restrict__ out) {
    __shared__ float s_f[8][KHOP];
    __shared__ float s_x[8][DIMIN];
    __shared__ float s_h[8][HID];
    __shared__ float s_g[8][HID];
    __shared__ float s_p[8][DIMPE];

    int wave = threadIdx.x >> 5;
    int lane = threadIdx.x & 31;
    int row  = blockIdx.x * 8 + wave;

    if (lane < KHOP) s_f[wave][lane] = feat[(size_t)row * KHOP + lane];
    s_x[wave][lane]      = x[(size_t)row * DIMIN + lane];
    s_x[wave][lane + 32] = x[(size_t)row * DIMIN + lane + 32];
    __syncthreads();

    // layer 1: [16] -> [64], 2 units per lane
    int u0 = 2 * lane, u1 = 2 * lane + 1;
    float h1a = b1[u0], h1b = b1[u1];
    for (int k = 0; k < KHOP; ++k) {
        float f = s_f[wave][k];
        h1a += f * W1[k * HID + u0];
        h1b += f * W1[k * HID + u1];
    }
    s_h[wave][u0] = fmaxf(h1a, 0.f);
    s_h[wave][u1] = fmaxf(h1b, 0.f);
    __syncthreads();

    // layer 2: [64] -> [64]
    float h2a = b2[u0], h2b = b2[u1];
    for (int k = 0; k < HID; ++k) {
        float f = s_h[wave][k];
        h2a += f * W2[k * HID + u0];
        h2b += f * W2[k * HID + u1];
    }
    s_g[wave][u0] = fmaxf(h2a, 0.f);
    s_g[wave][u1] = fmaxf(h2b, 0.f);
    __syncthreads();

    // layer 3: [64] -> [28]
    float h3 = 0.f;
    if (lane < DIMPE) {
        h3 = b3[lane];
        for (int k = 0; k < HID; ++k) h3 += s_g[wave][k] * W3[k * DIMPE + lane];
        h3 = fmaxf(h3, 0.f);
        s_p[wave][lane] = h3;
    }
    __syncthreads();

    // LayerNorm over 28 + write pos_enc part
    if (lane < DIMPE) {
        float mu = 0.f;
        for (int j = 0; j < DIMPE; ++j) mu += s_p[wave][j];
        mu *= (1.0f / DIMPE);
        float var = 0.f;
        for (int j = 0; j < DIMPE; ++j) { float d = s_p[wave][j] - mu; var += d * d; }
        var *= (1.0f / DIMPE);
        float pe = (h3 - mu) * rsqrtf(var + LN_EPS) * gamma[lane] + beta[lane];
        out[(size_t)row * (DIMX + DIMPE) + DIMX + lane] = pe;
    }

    // hx = x @ Wx + bx  -> first 100 columns
    for (int u = lane; u < DIMX; u += 32) {
        float v = bx[u];
        for (int k = 0; k < DIMIN; ++k) v += s_x[wave][k] * Wx[k * DIMX + u];
        out[(size_t)row * (DIMX + DIMPE) + u] = v;
    }
}

// ---------------------------------------------------------------- launcher
extern "C" void kernel_launch(void* const* d_in, const int* in_sizes, int n_in,
                              void* d_out, int out_size, void* d_ws, size_t ws_size,
                              hipStream_t stream) {
    const float* x     = (const float*)d_in[0];
    const int*   ei    = (const int*)d_in[1];
    const int*   frag  = (const int*)d_in[2];
    const float* alpha = (const float*)d_in[3];
    const float* Wx    = (const float*)d_in[4];
    const float* bx    = (const float*)d_in[5];
    const float* W1    = (const float*)d_in[6];
    const float* b1    = (const float*)d_in[7];
    const float* W2    = (const float*)d_in[8];
    const float* b2    = (const float*)d_in[9];
    const float* W3    = (const float*)d_in[10];
    const float* b3    = (const float*)d_in[11];
    const float* gamma = (const float*)d_in[12];
    const float* beta  = (const float*)d_in[13];
    float* out = (float*)d_out;
    float* ws  = (float*)d_ws;

    const size_t MAT = (size_t)G_ * N_ * N_;   // 4,194,304 floats (16 MB)
    float* P    = ws;
    float* Bf1  = ws + MAT;
    float* Bf2  = ws + 2 * MAT;
    float* feat = ws + 3 * MAT;                // 8192*16 floats

    // build P = D^-1 A
    zero_kernel<<<MAT / (4 * 256), 256, 0, stream>>>((float4*)P);
    scatter_kernel<<<(G_ * E_ + 255) / 256, 256, 0, stream>>>(ei, P);
    rownorm_kernel<<<(G_ * N_) / 256, 256, 0, stream>>>(P);

    // hop 0 (identity) and hop 1 (P itself)
    feat0_kernel<<<(G_ * N_) / 256, 256, 0, stream>>>(alpha, feat);
    reduce_kernel<<<(G_ * N_) / 256, 256, 0, stream>>>(P, frag, alpha, feat, 1);

    // power chain: P^k = P^(k-1) @ P, k = 2..15
    const float* cur = P;
    float* nxt = Bf1;
    float* alt = Bf2;
    for (int k = 2; k < KHOP; ++k) {
        gemm_pow_kernel<<<G_ * 16, 128, 0, stream>>>(cur, P, nxt);
        reduce_kernel<<<(G_ * N_) / 256, 256, 0, stream>>>(nxt, frag, alpha, feat, k);
        cur = nxt;
        float* t = nxt == Bf1 ? Bf2 : Bf1;
        nxt = t;
        alt = (float*)cur;
        (void)alt;
    }

    // fused MLP + LayerNorm + expand_x + concat
    mlp_out_kernel<<<(G_ * N_) / 8, 256, 0, stream>>>(
        feat, x, W1, b1, W2, b2, W3, b3, gamma, beta, Wx, bx, out);
}